// SymbolicEmbeddingsGumbel_38474317037595
// MI455X (gfx1250) — compile-verified
//
#include <hip/hip_runtime.h>
#include <stdint.h>

// Problem constants (from the reference): B=32, S=512, P=8, K=256, D=64
#define B_ 32
#define S_ 512
#define P_ 8
#define K_ 256
#define D_ 64
#define NROWS  (B_ * S_ * P_)   // 131072 (b,s,p) rows
#define NPAIRS (NROWS / 2)      // 65536 — row r pairs with r+65536 (shared threefry call)
#define HALF_N 16777216u        // (B*S*P*K)/2 = 2^24 : JAX threefry counter-half offset

#if defined(__has_builtin)
#  if __has_builtin(__builtin_amdgcn_tensor_load_to_lds) && __has_builtin(__builtin_amdgcn_s_wait_tensorcnt)
#    define USE_TDM 1
#  endif
#endif
#ifndef USE_TDM
#  define USE_TDM 0
#endif

typedef unsigned int uint32x4 __attribute__((ext_vector_type(4)));
typedef int          int32x8  __attribute__((ext_vector_type(8)));
typedef int          int32x4  __attribute__((ext_vector_type(4)));

// Exact JAX Threefry-2x32-20 with key = jax.random.key(42) -> (k0,k1) = (0,42).
__device__ __forceinline__ void threefry2x32_0_42(uint32_t c0, uint32_t c1,
                                                  uint32_t& o0, uint32_t& o1) {
  const uint32_t k0 = 0u;
  const uint32_t k1 = 42u;
  const uint32_t k2 = 0u ^ 42u ^ 0x1BD11BDAu;
  uint32_t x0 = c0 + k0, x1 = c1 + k1;
#define TF_R(r) { x0 += x1; x1 = (x1 << (r)) | (x1 >> (32 - (r))); x1 ^= x0; }
  TF_R(13) TF_R(15) TF_R(26) TF_R(6)
  x0 += k1; x1 += k2 + 1u;
  TF_R(17) TF_R(29) TF_R(16) TF_R(24)
  x0 += k2; x1 += k0 + 2u;
  TF_R(13) TF_R(15) TF_R(26) TF_R(6)
  x0 += k0; x1 += k1 + 3u;
  TF_R(17) TF_R(29) TF_R(16) TF_R(24)
  x0 += k1; x1 += k2 + 4u;
  TF_R(13) TF_R(15) TF_R(26) TF_R(6)
  x0 += k2; x1 += k0 + 5u;
#undef TF_R
  o0 = x0; o1 = x1;
}

// JAX: u = max(tiny, bitcast(bits>>9 | 0x3f800000) - 1) ; g = -log(-log(u))
__device__ __forceinline__ float bits_to_gumbel(uint32_t bits) {
  const float tiny = 1.17549435082228750797e-38f; // finfo(f32).tiny
  float f = __uint_as_float((bits >> 9) | 0x3F800000u) - 1.0f; // [0,1)
  float u = fmaxf(f, tiny); // == max(tiny, f*(1-tiny)+tiny) in f32
  return -__logf(-__logf(u));
}

__global__ __launch_bounds__(256) void sym_gumbel_kernel(
    const int*   __restrict__ inputs,       // [B*S] int32 categories
    const float* __restrict__ pattern_map,  // [N_CAT, P, K]
    const float* __restrict__ symbols,      // [K, D]  (64 KB)
    const float* __restrict__ tau_p,        // [1]
    float*       __restrict__ out)          // [B*S, P*D]
{
  __shared__ __align__(16) float lds_symbols[K_ * D_];  // 64 KB at LDS offset 0

  const int tid  = threadIdx.x;
  const int wave = tid >> 5;
  const int lane = tid & 31;

#if USE_TDM
  // ---- Phase 0: one TDM DMA of the whole symbols table into LDS (wave 0 only).
  // 2D descriptor: dim0 = D = 64 f32 (contiguous), dim1 = K = 256 rows; tile == tensor.
  if (wave == 0) {
    const uint64_t ga = (uint64_t)(uintptr_t)symbols;
    uint32x4 g0 = {
        1u,                                   // count=1, is_restore=0, gather_mode=0
        0u,                                   // lds_addr = 0 (lds_symbols is sole LDS alloc)
        (uint32_t)(ga & 0xFFFFFFFFu),         // global_addr[31:0]
        (uint32_t)((ga >> 32) & 0x01FFFFFFu)  // global_addr[56:32]
            | 0x80000000u };                  // type = 2 ("image")
    int32x8 g1 = {
        (int)0x00020000u,        // wg_mask=0, data_size=2 (4B), no barrier/iter/pad
        0, 0, 0, 0, 0, 0, 0 };
    g1[1] = (int)(((uint32_t)D_) << 16);        // bits[63:48]  = tensor_dim0 = 64
    g1[2] = (int)(((uint32_t)K_) << 16);        // bits[95:80]  = tensor_dim1 = 256
    g1[3] = (int)(((uint32_t)D_) << 16);        // bits[127:112]= tile_dim0   = 64
    g1[4] = (int)K_;                            // bits[143:128]= tile_dim1   = 256 (tile_dim2=0)
    g1[5] = (int)D_;                            // bits[191:160]= tensor_dim0_stride = 64
    g1[6] = (int)(((uint32_t)(K_ * D_)) << 16); // bits[223:208]= tensor_dim1_stride.lo16
    g1[7] = 0;
    int32x4 g2 = { 0, 0, 0, 0 };
    int32x4 g3 = { 0, 0, 0, 0 };
#if __clang_major__ >= 23
    int32x8 gpad = { 0, 0, 0, 0, 0, 0, 0, 0 };
    __builtin_amdgcn_tensor_load_to_lds(g0, g1, g2, g3, gpad, 0);
#else
    __builtin_amdgcn_tensor_load_to_lds(g0, g1, g2, g3, 0);
#endif
  }
  // CRITICAL: the TDM write to LDS is invisible to the optimizer. Without this,
  // LLVM proves lds_symbols has no stores, folds the gather loads to undef, and
  // DCEs the entire kernel (observed in round 1). This asm claims to write LDS.
  {
    float* lp = &lds_symbols[0];
    asm volatile("" : "+v"(lp) : : "memory");
  }
#else
  // Fallback: cooperative fill of symbols into LDS.
  for (int i = tid; i < K_ * D_; i += 256) lds_symbols[i] = symbols[i];
#endif

  // ---- Phase 1: Gumbel + argmax for a pair of rows per wave (shares threefry halves).
  const float tau = tau_p[0];
  const float sgn = (tau < 0.0f) ? -1.0f : 1.0f; // argmax((e+g)/tau): sign flips if tau<0

  const int pair = blockIdx.x * 8 + wave;  // 0 .. NPAIRS-1 (grid sized exactly)
  const int rowA = pair;                   // linear gumbel idx < 2^24
  const int rowB = pair + NPAIRS;          // its counter-partner (idx + 2^24)

  const int bsA = rowA >> 3, pA = rowA & 7;
  const int bsB = rowB >> 3, pB = rowB & 7;
  const int catA = inputs[bsA];
  const int catB = inputs[bsB];
  const float* eA = pattern_map + (size_t)catA * (P_ * K_) + (size_t)pA * K_;
  const float* eB = pattern_map + (size_t)catB * (P_ * K_) + (size_t)pB * K_;
  const uint32_t baseA = (uint32_t)rowA * (uint32_t)K_;

  float bvA = -__builtin_inff(); int biA = 0;
  float bvB = -__builtin_inff(); int biB = 0;

#pragma unroll
  for (int j = 0; j < 2; ++j) {
    const int k0 = (j * 32 + lane) * 4;               // wave reads contiguous 512B
    const float4 ea = *(const float4*)(eA + k0);
    const float4 eb = *(const float4*)(eB + k0);
#pragma unroll
    for (int t = 0; t < 4; ++t) {
      const int k = k0 + t;
      uint32_t r0, r1;
      threefry2x32_0_42(baseA + (uint32_t)k, baseA + (uint32_t)k + HALF_N, r0, r1);
      const float va = sgn * (((const float*)&ea)[t] + bits_to_gumbel(r0));
      const float vb = sgn * (((const float*)&eb)[t] + bits_to_gumbel(r1));
      if (va > bvA) { bvA = va; biA = k; }   // strict '>' keeps earliest k (JAX argmax)
      if (vb > bvB) { bvB = vb; biB = k; }
    }
  }

  // Wave-level argmax reduction (wave32), lower index wins ties.
#pragma unroll
  for (int off = 16; off > 0; off >>= 1) {
    float ov = __shfl_xor(bvA, off, 32);
    int   oi = __shfl_xor(biA, off, 32);
    if (ov > bvA || (ov == bvA && oi < biA)) { bvA = ov; biA = oi; }
    ov = __shfl_xor(bvB, off, 32);
    oi = __shfl_xor(biB, off, 32);
    if (ov > bvB || (ov == bvB && oi < biB)) { bvB = ov; biB = oi; }
  }

#if USE_TDM
  if (wave == 0) __builtin_amdgcn_s_wait_tensorcnt(0); // wave0 owns the TENSORcnt
#endif
  __syncthreads();  // releases all waves only after the DMA (or fill) is complete

  // ---- Phase 2: hard one-hot @ symbols == row gather from LDS. 2 floats/lane.
  const float2 sa = *(const float2*)&lds_symbols[biA * D_ + lane * 2];
  const float2 sb = *(const float2*)&lds_symbols[biB * D_ + lane * 2];
  *(float2*)(out + (size_t)rowA * D_ + lane * 2) = sa;
  *(float2*)(out + (size_t)rowB * D_ + lane * 2) = sb;
}

extern "C" void kernel_launch(void* const* d_in, const int* in_sizes, int n_in,
                              void* d_out, int out_size, void* d_ws, size_t ws_size,
                              hipStream_t stream) {
  (void)in_sizes; (void)n_in; (void)out_size; (void)d_ws; (void)ws_size;
  const int*   inputs      = (const int*)  d_in[0];
  const float* pattern_map = (const float*)d_in[1];
  const float* symbols     = (const float*)d_in[2];
  const float* tau         = (const float*)d_in[3];
  float*       out         = (float*)d_out;

  dim3 grid(NPAIRS / 8);  // 8192 blocks x 256 threads = 8 waves/block, 1 row-pair/wave
  sym_gumbel_kernel<<<grid, dim3(256), 0, stream>>>(inputs, pattern_map, symbols, tau, out);
}